// SinkhornDistance_11682311045540
// MI455X (gfx1250) — compile-verified
//
#include <hip/hip_runtime.h>
#include <math.h>

// Problem constants (fixed by the reference)
#define BATCH 8
#define NPTS 2048          // N == M
#define DFEAT 32
#define EPSV 0.1f
#define INV_EPS 10.0f
#define MAX_ITER 100
#define THRESHV 0.1f
#define SEG 8              // row-segments for the column pass

typedef __attribute__((ext_vector_type(2))) float v2f;
typedef __attribute__((ext_vector_type(8))) float v8f;

// ---- online logsumexp merge: (m,s) <- (m,s) (+) (mo,so) -------------------
__device__ __forceinline__ void lse_merge(float& m, float& s, float mo, float so) {
  float mn = fmaxf(m, mo);
  s = s * __expf(m - mn) + so * __expf(mo - mn);
  m = mn;
}

// ---- init: zero u, v, done ------------------------------------------------
__global__ void init_kernel(float* __restrict__ u, float* __restrict__ v, int* done) {
  int idx = blockIdx.x * 256 + threadIdx.x;      // 64 blocks -> 16384 threads
  u[idx] = 0.0f;
  v[idx] = 0.0f;
  if (idx == 0) *done = 0;
}

// ---- squared norms of each point ------------------------------------------
__global__ void norms_kernel(const float* __restrict__ x, const float* __restrict__ y,
                             float* __restrict__ xn, float* __restrict__ yn) {
  int p = blockIdx.x * 256 + threadIdx.x;        // 0..16383 (b*N+i)
  const float* xr = x + (size_t)p * DFEAT;
  const float* yr = y + (size_t)p * DFEAT;
  float sx = 0.0f, sy = 0.0f;
#pragma unroll
  for (int k = 0; k < DFEAT; ++k) {
    sx = fmaf(xr[k], xr[k], sx);
    sy = fmaf(yr[k], yr[k], sy);
  }
  xn[p] = sx;
  yn[p] = sy;
}

// ---- C = xnorm + ynorm - 2 * x . y^T  via V_WMMA_F32_16X16X4_F32 ----------
// grid: (NPTS/128, NPTS/16, BATCH), block: 256 (8 waves, one 16x16 tile each)
__global__ void gemm_C_kernel(const float* __restrict__ x, const float* __restrict__ y,
                              const float* __restrict__ xn, const float* __restrict__ yn,
                              float* __restrict__ C) {
  const int b = blockIdx.z;
  const int i_tile = blockIdx.y;                     // 0..127
  const int wave = threadIdx.x >> 5;                 // 0..7
  const int lane = threadIdx.x & 31;
  const int j_tile = blockIdx.x * 8 + wave;          // 0..127
  const int m = lane & 15;                           // A-row / B-col within tile
  const int khalf = (lane >> 4) * 2;                 // 0 or 2 (K pair per half-wave)

  const float* xrow = x + ((size_t)b * NPTS + (size_t)i_tile * 16 + m) * DFEAT;
  const float* yrow = y + ((size_t)b * NPTS + (size_t)j_tile * 16 + m) * DFEAT;

  v8f acc = {};
#pragma unroll
  for (int kk = 0; kk < DFEAT; kk += 4) {
    v2f afrag = *(const v2f*)(xrow + kk + khalf);    // A[m][k],A[m][k+1]
    v2f bfrag = *(const v2f*)(yrow + kk + khalf);    // B(=Y^T)[k][n] = y[n][k]
    acc = __builtin_amdgcn_wmma_f32_16x16x4_f32(
        /*neg_a=*/false, afrag, /*neg_b=*/false, bfrag,
        /*c_mod=*/(short)0, acc, /*reuse_a=*/false, /*reuse_b=*/false);
  }

  // D layout: VGPR r, lane t -> row r + (t>=16?8:0), col t&15
  const int col = lane & 15;
  const int rowoff = (lane >> 4) * 8;
  const float ynv = yn[(size_t)b * NPTS + (size_t)j_tile * 16 + col];
  float* Ctile = C + ((size_t)b * NPTS + (size_t)i_tile * 16) * NPTS + (size_t)j_tile * 16;
#pragma unroll
  for (int r = 0; r < 8; ++r) {
    const int row = r + rowoff;
    const float xv = xn[(size_t)b * NPTS + (size_t)i_tile * 16 + row];
    Ctile[(size_t)row * NPTS + col] = xv + ynv - 2.0f * acc[r];
  }
}

// ---- row pass: u_new[b,i] = eps*(log_mu - lse_j((v_j - C_ij)/eps)) --------
// one wave per row; grid: BATCH*NPTS/8 blocks of 256
__global__ void row_pass_kernel(const float* __restrict__ C, const float* __restrict__ v,
                                float* __restrict__ u, float* __restrict__ row_err,
                                const int* __restrict__ done, float log_mu) {
  if (*done) return;
  const int lane = threadIdx.x & 31;
  const int wave = threadIdx.x >> 5;
  const int row = blockIdx.x * 8 + wave;             // b*NPTS + i
  const int b = row >> 11;
  const float* Crow = C + (size_t)row * NPTS;
  const float* vb = v + (size_t)b * NPTS;

  float m = -INFINITY, s = 0.0f;
#pragma unroll
  for (int c = 0; c < 8; ++c) {                      // 8 chunks of 8 elems/lane
    const int j0 = c * 256 + lane * 4;               // coalesced b128 loads
    float4 ca = *(const float4*)(Crow + j0);
    float4 cb = *(const float4*)(Crow + j0 + 128);
    float4 va = *(const float4*)(vb + j0);
    float4 vv = *(const float4*)(vb + j0 + 128);
    __builtin_prefetch(Crow + j0 + 512, 0, 0);
    float z[8];
    z[0] = (va.x - ca.x) * INV_EPS; z[1] = (va.y - ca.y) * INV_EPS;
    z[2] = (va.z - ca.z) * INV_EPS; z[3] = (va.w - ca.w) * INV_EPS;
    z[4] = (vv.x - cb.x) * INV_EPS; z[5] = (vv.y - cb.y) * INV_EPS;
    z[6] = (vv.z - cb.z) * INV_EPS; z[7] = (vv.w - cb.w) * INV_EPS;
    float cm = z[0];
#pragma unroll
    for (int k = 1; k < 8; ++k) cm = fmaxf(cm, z[k]);
    float cs = 0.0f;
#pragma unroll
    for (int k = 0; k < 8; ++k) cs += __expf(z[k] - cm);
    lse_merge(m, s, cm, cs);
  }
  // wave32 butterfly reduction of (m,s)
#pragma unroll
  for (int off = 16; off > 0; off >>= 1) {
    float mo = __shfl_xor(m, off);
    float so = __shfl_xor(s, off);
    lse_merge(m, s, mo, so);
  }
  if (lane == 0) {
    const float u_new = EPSV * (log_mu - (m + __logf(s)));
    row_err[row] = fabsf(u_new - u[row]);
    u[row] = u_new;
  }
}

// ---- column pass, stage 1: per-segment partial (m,s) per column ----------
// grid: (NPTS/256, SEG, BATCH), block 256; thread owns one column
__global__ void col_partial_kernel(const float* __restrict__ C, const float* __restrict__ u,
                                   float* __restrict__ pm, float* __restrict__ ps,
                                   const int* __restrict__ done) {
  if (*done) return;
  const int b = blockIdx.z;
  const int seg = blockIdx.y;
  const int j = blockIdx.x * 256 + threadIdx.x;
  const float* Cb = C + (size_t)b * NPTS * NPTS;
  const float* ub = u + (size_t)b * NPTS;
  const int i0 = seg * (NPTS / SEG);

  float m = -INFINITY, s = 0.0f;
  for (int c = 0; c < (NPTS / SEG) / 8; ++c) {       // chunks of 8 rows
    const int ib = i0 + c * 8;
    __builtin_prefetch(Cb + (size_t)(ib + 8) * NPTS + j, 0, 0);
    float z[8];
#pragma unroll
    for (int r = 0; r < 8; ++r)
      z[r] = (ub[ib + r] - Cb[(size_t)(ib + r) * NPTS + j]) * INV_EPS;
    float cm = z[0];
#pragma unroll
    for (int k = 1; k < 8; ++k) cm = fmaxf(cm, z[k]);
    float cs = 0.0f;
#pragma unroll
    for (int k = 0; k < 8; ++k) cs += __expf(z[k] - cm);
    lse_merge(m, s, cm, cs);
  }
  const size_t idx = ((size_t)seg * BATCH + b) * NPTS + j;
  pm[idx] = m;
  ps[idx] = s;
}

// ---- column pass, stage 2: combine SEG partials, update v ----------------
__global__ void col_combine_kernel(const float* __restrict__ pm, const float* __restrict__ ps,
                                   float* __restrict__ v, const int* __restrict__ done,
                                   float log_nu) {
  if (*done) return;
  const int gid = blockIdx.x * 256 + threadIdx.x;    // b*NPTS + j
  const int b = gid >> 11;
  const int j = gid & (NPTS - 1);
  float m = -INFINITY, s = 0.0f;
#pragma unroll
  for (int seg = 0; seg < SEG; ++seg) {
    const size_t idx = ((size_t)seg * BATCH + b) * NPTS + j;
    lse_merge(m, s, pm[idx], ps[idx]);
  }
  v[gid] = EPSV * (log_nu - (m + __logf(s)));
}

// ---- err = mean_b sum_i |du| ; sticky done flag (deterministic reduce) ----
__global__ void err_reduce_kernel(const float* __restrict__ row_err, int* done) {
  if (*done) return;                                  // uniform branch
  __shared__ float sm[256];
  float s = 0.0f;
  for (int k = 0; k < (BATCH * NPTS) / 256; ++k)
    s += row_err[threadIdx.x + 256 * k];
  sm[threadIdx.x] = s;
  __syncthreads();
  for (int off = 128; off > 0; off >>= 1) {
    if (threadIdx.x < off) sm[threadIdx.x] += sm[threadIdx.x + off];
    __syncthreads();
  }
  if (threadIdx.x == 0) {
    if ((sm[0] / (float)BATCH) < THRESHV) *done = 1;
  }
}

// ---- pi = exp((u_i + v_j - C)/eps); per-row partial of sum(pi*C) ----------
__global__ void pi_cost_kernel(const float* __restrict__ C, const float* __restrict__ u,
                               const float* __restrict__ v, float* __restrict__ pi,
                               float* __restrict__ part) {
  const int row = blockIdx.x;                        // b*NPTS + i
  const int b = row >> 11;
  const float uv = u[row];
  const float* Crow = C + (size_t)row * NPTS;
  const float* vb = v + (size_t)b * NPTS;
  float* prow = pi + (size_t)row * NPTS;
  float acc = 0.0f;
#pragma unroll
  for (int k = 0; k < NPTS / 256; ++k) {
    const int j = threadIdx.x + 256 * k;
    const float cv = Crow[j];
    const float p = __expf((uv + vb[j] - cv) * INV_EPS);
    prow[j] = p;
    acc = fmaf(p, cv, acc);
  }
  __shared__ float sm[256];
  sm[threadIdx.x] = acc;
  __syncthreads();
  for (int off = 128; off > 0; off >>= 1) {
    if (threadIdx.x < off) sm[threadIdx.x] += sm[threadIdx.x + off];
    __syncthreads();
  }
  if (threadIdx.x == 0) part[row] = sm[0];
}

__global__ void cost_final_kernel(const float* __restrict__ part, float* __restrict__ cost) {
  const int b = blockIdx.x;
  __shared__ float sm[256];
  float s = 0.0f;
  for (int k = 0; k < NPTS / 256; ++k)
    s += part[(size_t)b * NPTS + threadIdx.x + 256 * k];
  sm[threadIdx.x] = s;
  __syncthreads();
  for (int off = 128; off > 0; off >>= 1) {
    if (threadIdx.x < off) sm[threadIdx.x] += sm[threadIdx.x + off];
    __syncthreads();
  }
  if (threadIdx.x == 0) cost[b] = sm[0];
}

extern "C" void kernel_launch(void* const* d_in, const int* in_sizes, int n_in,
                              void* d_out, int out_size, void* d_ws, size_t ws_size,
                              hipStream_t stream) {
  const float* x = (const float*)d_in[0];
  const float* y = (const float*)d_in[1];

  // d_out layout: cost[8] | pi[8*2048*2048] | C[8*2048*2048]
  float* out = (float*)d_out;
  float* cost = out;
  float* pi = out + BATCH;
  float* C = out + BATCH + (size_t)BATCH * NPTS * NPTS;

  // workspace layout (floats) — ~1.45 MB total
  float* ws = (float*)d_ws;
  float* u = ws;                                  // 16384
  float* v = u + BATCH * NPTS;                    // 16384
  float* xn = v + BATCH * NPTS;                   // 16384
  float* yn = xn + BATCH * NPTS;                  // 16384
  float* row_err = yn + BATCH * NPTS;             // 16384
  float* pm = row_err + BATCH * NPTS;             // SEG*8*2048 = 131072
  float* ps = pm + (size_t)SEG * BATCH * NPTS;    // 131072
  float* part = ps + (size_t)SEG * BATCH * NPTS;  // 16384
  int* done = (int*)(part + BATCH * NPTS);        // 1

  const float log_mu = logf(1.0f / (float)NPTS + 1e-8f);
  const float log_nu = logf(1.0f / (float)NPTS + 1e-8f);

  init_kernel<<<(BATCH * NPTS) / 256, 256, 0, stream>>>(u, v, done);
  norms_kernel<<<(BATCH * NPTS) / 256, 256, 0, stream>>>(x, y, xn, yn);
  gemm_C_kernel<<<dim3(NPTS / 128, NPTS / 16, BATCH), 256, 0, stream>>>(x, y, xn, yn, C);

  for (int it = 0; it < MAX_ITER; ++it) {
    row_pass_kernel<<<(BATCH * NPTS) / 8, 256, 0, stream>>>(C, v, u, row_err, done, log_mu);
    col_partial_kernel<<<dim3(NPTS / 256, SEG, BATCH), 256, 0, stream>>>(C, u, pm, ps, done);
    col_combine_kernel<<<(BATCH * NPTS) / 256, 256, 0, stream>>>(pm, ps, v, done, log_nu);
    err_reduce_kernel<<<1, 256, 0, stream>>>(row_err, done);
  }

  pi_cost_kernel<<<BATCH * NPTS, 256, 0, stream>>>(C, u, v, pi, part);
  cost_final_kernel<<<BATCH, 256, 0, stream>>>(part, cost);
}